// LinkConvInPillar2_44092134261327
// MI455X (gfx1250) — compile-verified
//
#include <hip/hip_runtime.h>
#include <hip/hip_bf16.h>

// MI455X (gfx1250) implementation.
// fp32 WMMA 16x16x4 for the [N,64]x[64,64] GEMM; memory-bound pipeline with
// L2-resident segment accumulators (51.2MB << 192MB L2).

typedef __attribute__((ext_vector_type(2))) float v2f;
typedef __attribute__((ext_vector_type(4))) float v4f;
typedef __attribute__((ext_vector_type(8))) float v8f;

#define NUM_SEG   100000
#define XPAD      68          // padded LDS row stride (floats): 16B-aligned rows, bank-conflict-free
#define EPS_BN    1e-3f
#define EPS_DIV   1e-4f

// ---------------------------------------------------------------------------
// Stage W (64x64) block-cooperatively and the wave's 16-row x tile into LDS,
// then run the 16-step K loop of V_WMMA_F32_16X16X4_F32 over 4 column tiles.
// A layout (f32 16x4): lane&15 = M row; lane>>4 selects K pair {0,1}/{2,3}.
// B layout (f32 4x16): lane&15 = N col; lane>>4 selects K pair.
// C/D layout: VGPR r holds row M = r + 8*(lane>>4), col N = 16t + (lane&15).
// ---------------------------------------------------------------------------
__device__ __forceinline__ void stage_and_gemm(
    const float* __restrict__ x, const float* __restrict__ Wm,
    float* sW, float* sXw, int rb_wave, int Nrows,
    int tid, int lane, v8f acc[4])
{
  // Stage W: 4096 floats, 256 threads, float4 each iter (fully coalesced).
  #pragma unroll
  for (int i = tid; i < 1024; i += 256) {
    int r  = i >> 4;            // W row (k)
    int c4 = (i & 15) << 2;     // W col
    v4f w = *(const v4f*)(Wm + (i << 2));
    *(v4f*)(sW + r * XPAD + c4) = w;
  }
  // Stage this wave's 16x64 x tile (rows rb_wave..rb_wave+15), zero-padded.
  #pragma unroll
  for (int i = lane; i < 256; i += 32) {
    int r   = i >> 4;
    int c4  = (i & 15) << 2;
    int row = rb_wave + r;
    v4f v = {0.f, 0.f, 0.f, 0.f};
    if (row < Nrows) v = *(const v4f*)(x + (size_t)row * 64 + c4);
    *(v4f*)(sXw + r * XPAD + c4) = v;
  }
  __syncthreads();

  const int L = lane & 15;
  const int h = lane >> 4;
  v8f z = {0.f,0.f,0.f,0.f,0.f,0.f,0.f,0.f};
  acc[0] = z; acc[1] = z; acc[2] = z; acc[3] = z;

  #pragma unroll
  for (int s = 0; s < 16; ++s) {
    int k = 4 * s + 2 * h;
    v2f a = *(const v2f*)(sXw + L * XPAD + k);   // ds_load_b64, conflict-free (stride 68)
    #pragma unroll
    for (int t = 0; t < 4; ++t) {
      v2f b;
      b.x = sW[(k    ) * XPAD + 16 * t + L];
      b.y = sW[(k + 1) * XPAD + 16 * t + L];
      acc[t] = __builtin_amdgcn_wmma_f32_16x16x4_f32(
          false, a, false, b, (short)0, acc[t], false, false);
    }
  }
}

// ---------------------------------------------------------------------------
// Pass 1: y = x@W (bias cancels in train-mode BN); per-channel sum / sumsq.
// ---------------------------------------------------------------------------
__global__ void k_gemm_stats1(const float* __restrict__ x, const float* __restrict__ Wm,
                              float* __restrict__ gsum, float* __restrict__ gsq, int Nrows)
{
  __shared__ float sW[64 * XPAD];
  __shared__ float sX[8][16 * XPAD];
  __shared__ float ssum[64], ssq[64];
  int tid = threadIdx.x, wave = tid >> 5, lane = tid & 31;
  if (tid < 64) { ssum[tid] = 0.f; ssq[tid] = 0.f; }
  int rb = blockIdx.x * 128 + wave * 16;
  v8f acc[4];
  stage_and_gemm(x, Wm, sW, &sX[wave][0], rb, Nrows, tid, lane, acc);

  int L = lane & 15, h = lane >> 4;
  #pragma unroll
  for (int t = 0; t < 4; ++t) {
    float s = 0.f, q = 0.f;
    #pragma unroll
    for (int r = 0; r < 8; ++r) {
      int row = rb + r + 8 * h;
      float y = acc[t][r];
      if (row < Nrows) { s += y; q += y * y; }
    }
    atomicAdd(&ssum[16 * t + L], s);
    atomicAdd(&ssq [16 * t + L], q);
  }
  __syncthreads();
  if (tid < 64) { atomicAdd(&gsum[tid], ssum[tid]); atomicAdd(&gsq[tid], ssq[tid]); }
}

// ---------------------------------------------------------------------------
// BN finalize: m = sum/N ; rstd = rsqrt(sumsq/N - m^2 + eps)   (biased var)
// ---------------------------------------------------------------------------
__global__ void k_finalize(const float* __restrict__ gsum, const float* __restrict__ gsq,
                           float* __restrict__ m, float* __restrict__ rstd,
                           float invN, float eps)
{
  int c = threadIdx.x;
  float mm = gsum[c] * invN;
  float v  = gsq[c] * invN - mm * mm;
  m[c] = mm;
  rstd[c] = rsqrtf(v + eps);
}

// ---------------------------------------------------------------------------
// Pass 2: recompute y -> feat=BN1(y); pos_w = floor(xyz)@W_pos + b_pos;
// atomic segment scatter of pos_w*feat and pos_w (L2-resident accumulators).
// ---------------------------------------------------------------------------
__global__ void k_scatter(const float* __restrict__ x, const float* __restrict__ Wm,
                          const float* __restrict__ xyz, const int* __restrict__ uinv,
                          const float* __restrict__ g1, const float* __restrict__ be1,
                          const float* __restrict__ Wpos, const float* __restrict__ bpos,
                          const float* __restrict__ m1, const float* __restrict__ r1,
                          float* __restrict__ feat_add, float* __restrict__ tmp_w, int Nrows)
{
  __shared__ float sW[64 * XPAD];
  __shared__ float sX[8][16 * XPAD];
  int tid = threadIdx.x, wave = tid >> 5, lane = tid & 31;
  int rb = blockIdx.x * 128 + wave * 16;
  v8f acc[4];
  stage_and_gemm(x, Wm, sW, &sX[wave][0], rb, Nrows, tid, lane, acc);

  int L = lane & 15, h = lane >> 4;
  float fx[8], fy[8], fz[8]; int uu[8];
  #pragma unroll
  for (int r = 0; r < 8; ++r) {
    int row = rb + r + 8 * h;
    int rc = row < Nrows ? row : 0;
    fx[r] = floorf(xyz[(size_t)rc * 3 + 0]);
    fy[r] = floorf(xyz[(size_t)rc * 3 + 1]);
    fz[r] = floorf(xyz[(size_t)rc * 3 + 2]);
    uu[r] = uinv[rc];
  }
  #pragma unroll
  for (int t = 0; t < 4; ++t) {
    int c = 16 * t + L;
    float sc = r1[c] * g1[c];
    float sh = be1[c] - m1[c] * sc;
    float w0 = Wpos[c], w1 = Wpos[64 + c], w2 = Wpos[128 + c], bp = bpos[c];
    #pragma unroll
    for (int r = 0; r < 8; ++r) {
      int row = rb + r + 8 * h;
      if (row < Nrows) {
        float pw   = fx[r] * w0 + fy[r] * w1 + fz[r] * w2 + bp;
        float feat = acc[t][r] * sc + sh;
        size_t off = (size_t)uu[r] * 64 + c;
        atomicAdd(&feat_add[off], pw * feat);
        atomicAdd(&tmp_w[off], pw);
      }
    }
  }
}

// ---------------------------------------------------------------------------
// Pass 3: stats of o = pos_w * (-feat_add[u]) / (tmp_w[u] + eps) for BN2.
// Thread layout: c = tid&63 (coalesced 128B gathers per wave), 4 row lanes.
// ---------------------------------------------------------------------------
__global__ void k_stats2(const float* __restrict__ xyz, const int* __restrict__ uinv,
                         const float* __restrict__ Wpos, const float* __restrict__ bpos,
                         const float* __restrict__ feat_add, const float* __restrict__ tmp_w,
                         float* __restrict__ gsum, float* __restrict__ gsq,
                         int Nrows, int rowsPerBlock)
{
  __shared__ float ssum[64], ssq[64];
  int tid = threadIdx.x;
  int c = tid & 63, r4 = tid >> 6;
  if (tid < 64) { ssum[tid] = 0.f; ssq[tid] = 0.f; }
  __syncthreads();
  float w0 = Wpos[c], w1 = Wpos[64 + c], w2 = Wpos[128 + c], bp = bpos[c];
  int base = blockIdx.x * rowsPerBlock;
  int end  = base + rowsPerBlock; if (end > Nrows) end = Nrows;
  float s = 0.f, q = 0.f;
  for (int row = base + r4; row < end; row += 4) {
    float fx = floorf(xyz[(size_t)row * 3 + 0]);
    float fy = floorf(xyz[(size_t)row * 3 + 1]);
    float fz = floorf(xyz[(size_t)row * 3 + 2]);
    int u = uinv[row];
    float pw = fx * w0 + fy * w1 + fz * w2 + bp;
    size_t off = (size_t)u * 64 + c;
    float fa = feat_add[off], tw = tmp_w[off];
    float o = pw * (-fa) / (tw + EPS_DIV);
    s += o; q += o * o;
  }
  atomicAdd(&ssum[c], s);
  atomicAdd(&ssq[c], q);
  __syncthreads();
  if (tid < 64) { atomicAdd(&gsum[tid], ssum[tid]); atomicAdd(&gsq[tid], ssq[tid]); }
}

// ---------------------------------------------------------------------------
// Pass 4: recompute feat; out = relu(BN2(o)) + feat.
// ---------------------------------------------------------------------------
__global__ void k_final(const float* __restrict__ x, const float* __restrict__ Wm,
                        const float* __restrict__ xyz, const int* __restrict__ uinv,
                        const float* __restrict__ g1, const float* __restrict__ be1,
                        const float* __restrict__ Wpos, const float* __restrict__ bpos,
                        const float* __restrict__ m1, const float* __restrict__ r1,
                        const float* __restrict__ m2, const float* __restrict__ r2,
                        const float* __restrict__ g2, const float* __restrict__ be2,
                        const float* __restrict__ feat_add, const float* __restrict__ tmp_w,
                        float* __restrict__ out, int Nrows)
{
  __shared__ float sW[64 * XPAD];
  __shared__ float sX[8][16 * XPAD];
  int tid = threadIdx.x, wave = tid >> 5, lane = tid & 31;
  int rb = blockIdx.x * 128 + wave * 16;
  v8f acc[4];
  stage_and_gemm(x, Wm, sW, &sX[wave][0], rb, Nrows, tid, lane, acc);

  int L = lane & 15, h = lane >> 4;
  float fx[8], fy[8], fz[8]; int uu[8];
  #pragma unroll
  for (int r = 0; r < 8; ++r) {
    int row = rb + r + 8 * h;
    int rc = row < Nrows ? row : 0;
    fx[r] = floorf(xyz[(size_t)rc * 3 + 0]);
    fy[r] = floorf(xyz[(size_t)rc * 3 + 1]);
    fz[r] = floorf(xyz[(size_t)rc * 3 + 2]);
    uu[r] = uinv[rc];
  }
  #pragma unroll
  for (int t = 0; t < 4; ++t) {
    int c = 16 * t + L;
    float sc1 = r1[c] * g1[c];
    float sh1 = be1[c] - m1[c] * sc1;
    float sc2 = r2[c] * g2[c];
    float sh2 = be2[c] - m2[c] * sc2;
    float w0 = Wpos[c], w1 = Wpos[64 + c], w2 = Wpos[128 + c], bp = bpos[c];
    #pragma unroll
    for (int r = 0; r < 8; ++r) {
      int row = rb + r + 8 * h;
      if (row < Nrows) {
        float pw   = fx[r] * w0 + fy[r] * w1 + fz[r] * w2 + bp;
        float feat = acc[t][r] * sc1 + sh1;
        size_t off = (size_t)uu[r] * 64 + c;
        float fa = feat_add[off], tw = tmp_w[off];
        float o  = pw * (-fa) / (tw + EPS_DIV);
        float bn = o * sc2 + sh2;
        out[(size_t)row * 64 + c] = fmaxf(bn, 0.f) + feat;
      }
    }
  }
}

// ---------------------------------------------------------------------------
extern "C" void kernel_launch(void* const* d_in, const int* in_sizes, int n_in,
                              void* d_out, int out_size, void* d_ws, size_t ws_size,
                              hipStream_t stream)
{
  const float* xyz  = (const float*)d_in[0];
  const float* feat = (const float*)d_in[1];
  const int*   uinv = (const int*)  d_in[2];
  const float* Wpre = (const float*)d_in[3];
  // d_in[4] = b_pre: unused (cancels inside training-mode BatchNorm)
  const float* g1   = (const float*)d_in[5];
  const float* be1  = (const float*)d_in[6];
  const float* Wpos = (const float*)d_in[7];
  const float* bpos = (const float*)d_in[8];
  const float* g2   = (const float*)d_in[9];
  const float* be2  = (const float*)d_in[10];
  float* out = (float*)d_out;
  int N = in_sizes[2];

  float* ws = (float*)d_ws;
  float *s1 = ws,       *q1 = ws + 64,  *m1 = ws + 128, *r1 = ws + 192;
  float *s2 = ws + 256, *q2 = ws + 320, *m2 = ws + 384, *r2 = ws + 448;
  float* fa = ws + 512;
  float* tw = fa + (size_t)NUM_SEG * 64;

  size_t zero_bytes = (512 + 2 * (size_t)NUM_SEG * 64) * sizeof(float);
  if (zero_bytes > ws_size) zero_bytes = ws_size;
  hipMemsetAsync(d_ws, 0, zero_bytes, stream);

  int nb = (N + 127) / 128;
  k_gemm_stats1<<<nb, 256, 0, stream>>>(feat, Wpre, s1, q1, N);
  k_finalize<<<1, 64, 0, stream>>>(s1, q1, m1, r1, 1.f / (float)N, EPS_BN);
  k_scatter<<<nb, 256, 0, stream>>>(feat, Wpre, xyz, uinv, g1, be1, Wpos, bpos,
                                    m1, r1, fa, tw, N);
  int rpb = 512;
  int nb2 = (N + rpb - 1) / rpb;
  k_stats2<<<nb2, 256, 0, stream>>>(xyz, uinv, Wpos, bpos, fa, tw, s2, q2, N, rpb);
  k_finalize<<<1, 64, 0, stream>>>(s2, q2, m2, r2, 1.f / (float)N, EPS_BN);
  k_final<<<nb, 256, 0, stream>>>(feat, Wpre, xyz, uinv, g1, be1, Wpos, bpos,
                                  m1, r1, m2, r2, g2, be2, fa, tw, out, N);
}